// AttentionDecoder_74440373174878
// MI455X (gfx1250) — compile-verified
//
#include <hip/hip_runtime.h>
#include <math.h>

typedef __attribute__((ext_vector_type(2))) float v2f;
typedef __attribute__((ext_vector_type(8))) float v8f;

#define HIDDEN 1024
#define VOCAB  128000
#define MAXLEN 10

// ---------------------------------------------------------------------------
// GEMV via V_WMMA_F32_16X16X4_F32.
// One wave computes 16 output rows: out[row0+m] = dot(W[row0+m,:], vec) + bias.
// A operand = 16x4 fp32 tile of W (every loaded byte unique -> no extra BW).
// B operand = vec broadcast into column N=0, zeros elsewhere.
// D column 0 lives in lane 0 (M=0..7) / lane 16 (M=8..15) per ISA 7.12.2.
// MODE 0: +bias.  MODE 1: relu(+bias).
// ---------------------------------------------------------------------------
template <int MODE>
__global__ void wmma_gemv_f32(const float* __restrict__ W,
                              const float* __restrict__ vec,
                              const float* __restrict__ bias,
                              float* __restrict__ out,
                              int K, int rows)
{
    extern __shared__ float sh[];          // K floats: RHS vector
    for (int i = threadIdx.x; i < K; i += blockDim.x) sh[i] = vec[i];
    __syncthreads();

    const int lane = threadIdx.x & 31;
    const int wave = threadIdx.x >> 5;
    const int tile = blockIdx.x * (blockDim.x >> 5) + wave;
    const int row0 = tile * 16;
    if (row0 >= rows) return;

    const int   m     = lane & 15;
    const int   kb    = (lane >> 4) * 2;                // 0 or 2
    const float bmask = (m == 0) ? 1.0f : 0.0f;         // only column N=0 carries vec
    const float* wrow = W + (size_t)(row0 + m) * (size_t)K + kb;

    v8f acc = {};
#pragma unroll 8
    for (int k0 = 0; k0 < K; k0 += 4) {
        v2f a, b;
        a.x = wrow[k0];                                 // -> global_load_b64
        a.y = wrow[k0 + 1];
        b.x = bmask * sh[k0 + kb];                      // -> ds_load, masked
        b.y = bmask * sh[k0 + kb + 1];
        acc = __builtin_amdgcn_wmma_f32_16x16x4_f32(
            /*neg_a=*/false, a, /*neg_b=*/false, b,
            /*c_mod=*/(short)0, acc, /*reuse_a=*/false, /*reuse_b=*/false);
    }

    if (lane == 0) {                                    // rows row0+0..7
#pragma unroll
        for (int i = 0; i < 8; ++i) {
            float r = acc[i] + bias[row0 + i];
            if (MODE == 1) r = fmaxf(r, 0.0f);
            out[row0 + i] = r;
        }
    } else if (lane == 16) {                            // rows row0+8..15
#pragma unroll
        for (int i = 0; i < 8; ++i) {
            float r = acc[i] + bias[row0 + 8 + i];
            if (MODE == 1) r = fmaxf(r, 0.0f);
            out[row0 + 8 + i] = r;
        }
    }
}

// ---------------------------------------------------------------------------
// Stage 1: embedding gather, additive-attention scores, softmax(10),
// attention_applied.  Single block of 256 threads (tiny: ~120KB traffic).
// Writes comb_in = [embedded | attention_applied] (2048 floats in ws)
// and attention_weights (10 floats, straight to d_out tail).
// ---------------------------------------------------------------------------
__global__ void attn_kernel(const int* __restrict__ input_,
                            const float* __restrict__ hidden,
                            const float* __restrict__ enc,
                            const float* __restrict__ embW,
                            const float* __restrict__ attnW,
                            const float* __restrict__ attnb,
                            float* __restrict__ comb_in,
                            float* __restrict__ attw_out)
{
    __shared__ float red[256];
    __shared__ float w[MAXLEN];
    const int tid = threadIdx.x;
    const int idx = input_[0];
    const float* erow = embW + (size_t)idx * HIDDEN;

    for (int j = tid; j < HIDDEN; j += 256) comb_in[j] = erow[j];   // embedded

    for (int i = 0; i < MAXLEN; ++i) {                 // scores over [emb|h0]
        float s = 0.0f;
        for (int j = tid; j < 2 * HIDDEN; j += 256) {
            float x = (j < HIDDEN) ? erow[j] : hidden[j - HIDDEN];
            s += attnW[i * 2 * HIDDEN + j] * x;
        }
        red[tid] = s; __syncthreads();
        for (int off = 128; off > 0; off >>= 1) {
            if (tid < off) red[tid] += red[tid + off];
            __syncthreads();
        }
        if (tid == 0) w[i] = red[0] + attnb[i];
        __syncthreads();
    }
    if (tid == 0) {                                    // softmax over 10
        float mx = w[0];
        for (int i = 1; i < MAXLEN; ++i) mx = fmaxf(mx, w[i]);
        float sm = 0.0f;
        for (int i = 0; i < MAXLEN; ++i) { w[i] = expf(w[i] - mx); sm += w[i]; }
        float inv = 1.0f / sm;
        for (int i = 0; i < MAXLEN; ++i) { w[i] *= inv; attw_out[i] = w[i]; }
    }
    __syncthreads();
    for (int j = tid; j < HIDDEN; j += 256) {          // weighted sum of enc
        float s = 0.0f;
#pragma unroll
        for (int i = 0; i < MAXLEN; ++i) s += w[i] * enc[i * HIDDEN + j];
        comb_in[HIDDEN + j] = s;
    }
}

// ---------------------------------------------------------------------------
// Stage 4: GRU gates (gi/gh already include their biases from the GEMV).
// ---------------------------------------------------------------------------
__global__ void gru_gate_kernel(const float* __restrict__ gi,
                                const float* __restrict__ gh,
                                const float* __restrict__ hidden,
                                float* __restrict__ h_ws,
                                float* __restrict__ h_out)
{
    int j = blockIdx.x * blockDim.x + threadIdx.x;
    if (j >= HIDDEN) return;
    float r = 1.0f / (1.0f + expf(-(gi[j]            + gh[j])));
    float z = 1.0f / (1.0f + expf(-(gi[HIDDEN + j]   + gh[HIDDEN + j])));
    float n = tanhf(gi[2 * HIDDEN + j] + r * gh[2 * HIDDEN + j]);
    float h = (1.0f - z) * n + z * hidden[j];
    h_ws[j] = h;
    h_out[j] = h;
}

// ---------------------------------------------------------------------------
// log_softmax over VOCAB: two-pass grid reductions over logits in d_out.
// ---------------------------------------------------------------------------
__global__ void reduce_max_partial(const float* __restrict__ x, float* __restrict__ part, int n)
{
    __shared__ float red[256];
    float m = -INFINITY;
    for (int i = blockIdx.x * blockDim.x + threadIdx.x; i < n; i += gridDim.x * blockDim.x)
        m = fmaxf(m, x[i]);
    red[threadIdx.x] = m; __syncthreads();
    for (int off = 128; off > 0; off >>= 1) {
        if (threadIdx.x < off) red[threadIdx.x] = fmaxf(red[threadIdx.x], red[threadIdx.x + off]);
        __syncthreads();
    }
    if (threadIdx.x == 0) part[blockIdx.x] = red[0];
}

__global__ void reduce_max_final(const float* __restrict__ part, int n, float* __restrict__ gmax)
{
    __shared__ float red[256];
    float m = -INFINITY;
    for (int i = threadIdx.x; i < n; i += 256) m = fmaxf(m, part[i]);
    red[threadIdx.x] = m; __syncthreads();
    for (int off = 128; off > 0; off >>= 1) {
        if (threadIdx.x < off) red[threadIdx.x] = fmaxf(red[threadIdx.x], red[threadIdx.x + off]);
        __syncthreads();
    }
    if (threadIdx.x == 0) *gmax = red[0];
}

__global__ void reduce_sumexp_partial(const float* __restrict__ x, const float* __restrict__ gmax,
                                      float* __restrict__ part, int n)
{
    __shared__ float red[256];
    float mx = *gmax;
    float s = 0.0f;
    for (int i = blockIdx.x * blockDim.x + threadIdx.x; i < n; i += gridDim.x * blockDim.x)
        s += expf(x[i] - mx);
    red[threadIdx.x] = s; __syncthreads();
    for (int off = 128; off > 0; off >>= 1) {
        if (threadIdx.x < off) red[threadIdx.x] += red[threadIdx.x + off];
        __syncthreads();
    }
    if (threadIdx.x == 0) part[blockIdx.x] = red[0];
}

__global__ void reduce_sum_final(const float* __restrict__ part, int n,
                                 const float* __restrict__ gmax, float* __restrict__ logZ)
{
    __shared__ float red[256];
    float s = 0.0f;
    for (int i = threadIdx.x; i < n; i += 256) s += part[i];
    red[threadIdx.x] = s; __syncthreads();
    for (int off = 128; off > 0; off >>= 1) {
        if (threadIdx.x < off) red[threadIdx.x] += red[threadIdx.x + off];
        __syncthreads();
    }
    if (threadIdx.x == 0) *logZ = *gmax + logf(red[0]);
}

__global__ void logsoftmax_write(float* __restrict__ out, const float* __restrict__ logZ, int n)
{
    float lz = *logZ;
    for (int i = blockIdx.x * blockDim.x + threadIdx.x; i < n; i += gridDim.x * blockDim.x)
        out[i] -= lz;
}

// ---------------------------------------------------------------------------
extern "C" void kernel_launch(void* const* d_in, const int* in_sizes, int n_in,
                              void* d_out, int out_size, void* d_ws, size_t ws_size,
                              hipStream_t stream)
{
    const int*   input_ = (const int*)  d_in[0];
    const float* hidden = (const float*)d_in[1];
    const float* enc    = (const float*)d_in[2];
    const float* embW   = (const float*)d_in[3];
    const float* attnW  = (const float*)d_in[4];
    const float* attnb  = (const float*)d_in[5];
    const float* combW  = (const float*)d_in[6];
    const float* combb  = (const float*)d_in[7];
    const float* Wih    = (const float*)d_in[8];
    const float* bih    = (const float*)d_in[9];
    const float* Whh    = (const float*)d_in[10];
    const float* bhh    = (const float*)d_in[11];
    const float* outW   = (const float*)d_in[12];
    const float* outb   = (const float*)d_in[13];

    float* out = (float*)d_out;      // [0,VOCAB): logits->logprobs ; [VOCAB,+H): h_new ; then 10 attw
    float* ws  = (float*)d_ws;
    float* comb_in = ws;             // 2048  [embedded | attention_applied]
    float* x       = ws + 2048;      // 1024
    float* gi      = ws + 3072;      // 3072
    float* gh      = ws + 6144;      // 3072
    float* h_ws    = ws + 9216;      // 1024
    float* pmax    = ws + 10240;     // 512
    float* psum    = ws + 10752;     // 512
    float* gmax    = ws + 11264;     // 1
    float* logZ    = ws + 11265;     // 1

    // 1) embedding + attention + applied  -> comb_in, attw -> d_out tail
    attn_kernel<<<1, 256, 0, stream>>>(input_, hidden, enc, embW, attnW, attnb,
                                       comb_in, out + VOCAB + HIDDEN);

    // 2) x = relu(comb_W @ comb_in + comb_b)   (1024 x 2048)
    wmma_gemv_f32<1><<<1024 / 16 / 8, 256, 2 * HIDDEN * sizeof(float), stream>>>(
        combW, comb_in, combb, x, 2 * HIDDEN, HIDDEN);

    // 3) gi = W_ih @ x + b_ih ; gh = W_hh @ h0 + b_hh   (3072 x 1024 each)
    wmma_gemv_f32<0><<<3072 / 16 / 8, 256, HIDDEN * sizeof(float), stream>>>(
        Wih, x, bih, gi, HIDDEN, 3 * HIDDEN);
    wmma_gemv_f32<0><<<3072 / 16 / 8, 256, HIDDEN * sizeof(float), stream>>>(
        Whh, hidden, bhh, gh, HIDDEN, 3 * HIDDEN);

    // 4) GRU gates -> h_new (ws + d_out)
    gru_gate_kernel<<<(HIDDEN + 255) / 256, 256, 0, stream>>>(gi, gh, hidden,
                                                              h_ws, out + VOCAB);

    // 5) logits = out_W @ h_new + out_b   (128000 x 1024, the 512MB stream)
    wmma_gemv_f32<0><<<VOCAB / 16 / 8, 256, HIDDEN * sizeof(float), stream>>>(
        outW, h_ws, outb, out, HIDDEN, VOCAB);

    // 6) log_softmax in place over d_out[0..VOCAB)
    reduce_max_partial   <<<512, 256, 0, stream>>>(out, pmax, VOCAB);
    reduce_max_final     <<<1,   256, 0, stream>>>(pmax, 512, gmax);
    reduce_sumexp_partial<<<512, 256, 0, stream>>>(out, gmax, psum, VOCAB);
    reduce_sum_final     <<<1,   256, 0, stream>>>(psum, 512, gmax, logZ);
    logsoftmax_write     <<<500, 256, 0, stream>>>(out, logZ, VOCAB);
}